// MRF_CVAE_23682449670910
// MI455X (gfx1250) — compile-verified
//
#include <hip/hip_runtime.h>
#include <hip/hip_bf16.h>
#include <stdint.h>

// ---------------- problem constants (match reference setup_inputs) ----------
#define P_PED   768
#define THIST   8
#define DH      256
#define NSAMP   20
#define STRIDEK 5
#define FUT     20
#define ZDIM    16
#define F2DIM   32
#define NSCENES 24
#define ROWS    (NSAMP * P_PED)      // 15360

typedef float v2f __attribute__((ext_vector_type(2)));
typedef float v8f __attribute__((ext_vector_type(8)));
typedef unsigned int u32x4 __attribute__((ext_vector_type(4)));
typedef int i32x8 __attribute__((ext_vector_type(8)));
typedef int i32x4 __attribute__((ext_vector_type(4)));

// ---------------- threefry2x32 (JAX-compatible), host + device -------------
__host__ __device__ static inline uint32_t rotl32_(uint32_t x, int d) {
  return (x << d) | (x >> (32 - d));
}
__host__ __device__ static inline void threefry2x32_(uint32_t k0, uint32_t k1,
                                                     uint32_t x0, uint32_t x1,
                                                     uint32_t* o0, uint32_t* o1) {
  uint32_t ks0 = k0, ks1 = k1, ks2 = 0x1BD11BDAu ^ k0 ^ k1;
  x0 += ks0; x1 += ks1;
#define TFR4(a,b,c,d) \
  x0 += x1; x1 = rotl32_(x1, a); x1 ^= x0; \
  x0 += x1; x1 = rotl32_(x1, b); x1 ^= x0; \
  x0 += x1; x1 = rotl32_(x1, c); x1 ^= x0; \
  x0 += x1; x1 = rotl32_(x1, d); x1 ^= x0;
  TFR4(13,15,26,6)  x0 += ks1; x1 += ks2 + 1u;
  TFR4(17,29,16,24) x0 += ks2; x1 += ks0 + 2u;
  TFR4(13,15,26,6)  x0 += ks0; x1 += ks1 + 3u;
  TFR4(17,29,16,24) x0 += ks1; x1 += ks2 + 4u;
  TFR4(13,15,26,6)  x0 += ks2; x1 += ks0 + 5u;
#undef TFR4
  *o0 = x0; *o1 = x1;
}

// host-side jax.random.split(key) -> (new_key, sub)
static void host_split_(uint32_t key[2], uint32_t sub[2]) {
  uint32_t a0, a1, b0, b1;
  threefry2x32_(key[0], key[1], 0u, 2u, &a0, &a1);
  threefry2x32_(key[0], key[1], 1u, 3u, &b0, &b1);
  key[0] = a0; key[1] = b0;
  sub[0] = a1; sub[1] = b1;
}

__device__ static inline float erfinvf_(float x) {
  float w = -logf((1.0f - x) * (1.0f + x));
  float p;
  if (w < 5.0f) {
    w -= 2.5f;
    p = 2.81022636e-08f;
    p = fmaf(p, w, 3.43273939e-07f);
    p = fmaf(p, w, -3.5233877e-06f);
    p = fmaf(p, w, -4.39150654e-06f);
    p = fmaf(p, w, 0.00021858087f);
    p = fmaf(p, w, -0.00125372503f);
    p = fmaf(p, w, -0.00417768164f);
    p = fmaf(p, w, 0.246640727f);
    p = fmaf(p, w, 1.50140941f);
  } else {
    w = sqrtf(w) - 3.0f;
    p = -0.000200214257f;
    p = fmaf(p, w, 0.000100950558f);
    p = fmaf(p, w, 0.00134934322f);
    p = fmaf(p, w, -0.00367342844f);
    p = fmaf(p, w, 0.00573950773f);
    p = fmaf(p, w, -0.0076224613f);
    p = fmaf(p, w, 0.00943887047f);
    p = fmaf(p, w, 1.00167406f);
    p = fmaf(p, w, 2.83297682f);
  }
  return p * x;
}

// z[i] = sigma * sqrt(2) * erfinv(uniform(-1,1)) using threefry(key, iota(n))
__global__ void k_normal(uint32_t k0, uint32_t k1, int n, float sigma,
                         float* __restrict__ out) {
  int j = blockIdx.x * blockDim.x + threadIdx.x;
  int half = n >> 1;
  if (j >= half) return;
  uint32_t o0, o1;
  threefry2x32_(k0, k1, (uint32_t)j, (uint32_t)(j + half), &o0, &o1);
  const float lo = -0.99999994f, hi = 1.0f;
  float f0 = __uint_as_float((o0 >> 9) | 0x3F800000u) - 1.0f;
  float f1 = __uint_as_float((o1 >> 9) | 0x3F800000u) - 1.0f;
  float u0 = fmaxf(lo, fmaf(f0, hi - lo, lo));
  float u1 = fmaxf(lo, fmaf(f1, hi - lo, lo));
  float s = sigma * 1.41421356237f;
  out[j]        = s * erfinvf_(u0);
  out[j + half] = s * erfinvf_(u1);
}

// ---------------- TDM: DMA one 64-row x 32-col f32 tile into LDS -----------
// D# built per CDNA5 ISA. kRem = K - k0: hardware
// zero-fills reads beyond tensor_dim0 (the K tail). pad_interval=4 (32 DW) +
// pad_amount=1 (2 DW) => LDS row stride 34 floats (matches As[.][64][34]).
__device__ static inline void tdm_load_tile(const float* gsrc, int ldaElems,
                                            int kRem, uint32_t ldsByteAddr) {
  const uint64_t ga = (uint64_t)(uintptr_t)gsrc;
  u32x4 g0;
  g0[0] = 1u;                                   // count=1 valid user descriptor
  g0[1] = ldsByteAddr;                          // lds_addr
  g0[2] = (uint32_t)ga;                         // global_addr[31:0]
  g0[3] = ((uint32_t)(ga >> 32) & 0x01FFFFFFu)  // global_addr[56:32]
          | 0x80000000u;                        // type=2 ("image")
  i32x8 g1;
  g1[0] = (int)((2u << 16)    // data_size = 4B
                | (1u << 20)  // pad_enable
                | (4u << 22)  // pad_interval: 32 DWORDs
                | (1u << 25)); // pad_amount: 2 DWORDs
  g1[1] = (int)(((uint32_t)kRem & 0xFFFFu) << 16);  // tensor_dim0[15:0] (<=512)
  g1[2] = (int)(64u << 16);   // tensor_dim0[31:16]=0 | tensor_dim1=64
  g1[3] = (int)(32u << 16);   // tensor_dim1[31:16]=0 | tile_dim0=32
  g1[4] = 64;                 // tile_dim1=64 | tile_dim2=0
  g1[5] = ldaElems;           // tensor_dim0_stride[31:0] (data_size units)
  g1[6] = 0;                  // stride0[47:32]=0 | stride1[15:0]=0
  g1[7] = 0;
  const i32x4 gz4 = {0, 0, 0, 0};               // 2D tensor: groups 2/3 unused
  const i32x8 gz8 = {0, 0, 0, 0, 0, 0, 0, 0};
  __builtin_amdgcn_tensor_load_to_lds(g0, g1, gz4, gz4, gz8, 0);
}

// ---------------- WMMA f32 GEMM: C = act(A@B + bias [+ res]) ---------------
// A[M,K], B[K,N] row-major. Block = 128 threads (4 waves), tile 64x64.
// A tile (64 x 32 K) is DMA'd into LDS by the Tensor Data Mover, double
// buffered so the next stage's DMA overlaps this stage's WMMAs. M%64==0.
#define KT 32
__global__ __launch_bounds__(128) void k_gemm(
    const float* __restrict__ A, int lda,
    const float* __restrict__ B, int ldb,
    const float* __restrict__ bias,
    const float* __restrict__ residual,
    float* __restrict__ C, int ldc,
    int M, int N, int K, int relu) {
  __shared__ float As[2][64][34];              // stride 34: 8B-aligned v2f rows,
                                               // conflict-free column reads
  const int tid  = threadIdx.x;
  const int wave = tid >> 5;
  const int lane = tid & 31;
  const int lh   = lane & 15;
  const int khi  = (lane >> 4) << 1;           // 0 (lanes 0-15) or 2 (16-31)
  const int colB = blockIdx.x * 64 + wave * 16;
  const int rowB = blockIdx.y * 64;
  const int col  = colB + lh;
  const bool cok = (col < N);
  const float* Bc = B + (cok ? col : (N - 1)); // clamped: always a valid address
  const float* Atile = A + (size_t)rowB * lda;

  v8f acc[4];
  acc[0] = (v8f)(0.0f); acc[1] = (v8f)(0.0f);
  acc[2] = (v8f)(0.0f); acc[3] = (v8f)(0.0f);

  const int nst = (K + KT - 1) / KT;
  if (wave == 0)
    tdm_load_tile(Atile, lda, K, (uint32_t)(uintptr_t)&As[0][0][0]);

  for (int st = 0; st < nst; ++st) {
    const int k0 = st * KT;
    if (wave == 0) __builtin_amdgcn_s_wait_tensorcnt(0);
    __syncthreads();                           // stage tile visible to all waves
    if (wave == 0 && st + 1 < nst)             // prefetch next stage via TDM
      tdm_load_tile(Atile + (k0 + KT), lda, K - (k0 + KT),
                    (uint32_t)(uintptr_t)&As[(st + 1) & 1][0][0]);
    const float (*Ab)[34] = As[st & 1];
#pragma unroll
    for (int j = 0; j < KT / 4; ++j) {
      const int kl = 4 * j + khi;              // local k index in LDS tile
      const int ka = k0 + kl;
      // clamp B addresses past K: A rows there are hardware-zero-filled,
      // so the garbage B values are multiplied by zero.
      const int k1 = (ka     < K) ? ka     : (K - 1);
      const int k2 = (ka + 1 < K) ? ka + 1 : (K - 1);
      v2f bv;
      bv.x = Bc[(size_t)k1 * ldb];
      bv.y = Bc[(size_t)k2 * ldb];
#pragma unroll
      for (int s = 0; s < 4; ++s) {
        const v2f av = *(const v2f*)&Ab[s * 16 + lh][kl];   // ds_load_b64
        acc[s] = __builtin_amdgcn_wmma_f32_16x16x4_f32(
            false, av, false, bv, (short)0, acc[s], false, false);
      }
    }
    __syncthreads();                           // all reads done before overwrite
  }

  if (!cok) return;
  const float bc = bias ? bias[col] : 0.0f;
#pragma unroll
  for (int s = 0; s < 4; ++s) {
#pragma unroll
    for (int i = 0; i < 8; ++i) {
      const int row = rowB + s * 16 + ((lane < 16) ? i : i + 8);
      float v = acc[s][i] + bc;
      if (residual) v += residual[(size_t)row * ldc + col];
      if (relu) v = fmaxf(v, 0.0f);
      C[(size_t)row * ldc + col] = v;
    }
  }
}

// ---------------- attention core: per-ped 8x8 softmax attention -------------
__global__ void k_attn(const float* __restrict__ q, const float* __restrict__ k,
                       const float* __restrict__ v, const float* __restrict__ mask,
                       float* __restrict__ attn_out) {
  const int p = blockIdx.x;
  const int tid = threadIdx.x;                  // 64 threads
  __shared__ float sc[THIST][THIST];
  {
    const int t = tid >> 3, s = tid & 7;
    const float* qr = q + (size_t)(p * THIST + t) * DH;
    const float* kr = k + (size_t)(p * THIST + s) * DH;
    float d = 0.0f;
    for (int i = 0; i < DH; ++i) d = fmaf(qr[i], kr[i], d);
    d *= 0.0625f;                               // 1/sqrt(256)
    sc[t][s] = (mask[p * THIST + s] > 0.0f) ? d : -1e9f;
  }
  __syncthreads();
  if (tid < THIST) {
    float mx = -1e30f;
    for (int j = 0; j < THIST; ++j) mx = fmaxf(mx, sc[tid][j]);
    float e[THIST], sum = 0.0f;
    for (int j = 0; j < THIST; ++j) { e[j] = __expf(sc[tid][j] - mx); sum += e[j]; }
    const float inv = 1.0f / sum;
    for (int j = 0; j < THIST; ++j) sc[tid][j] = e[j] * inv;
  }
  __syncthreads();
  const int t = tid >> 3, d0 = tid & 7;
  for (int j = 0; j < 32; ++j) {
    const int d = d0 + j * 8;
    float a = 0.0f;
    for (int s = 0; s < THIST; ++s)
      a = fmaf(sc[t][s], v[(size_t)(p * THIST + s) * DH + d], a);
    attn_out[(size_t)(p * THIST + t) * DH + d] = a;
  }
}

// masked mean pool over T
__global__ void k_pool(const float* __restrict__ h2, const float* __restrict__ mask,
                       float* __restrict__ hist) {
  const int p = blockIdx.x, d = threadIdx.x;    // 256 threads
  float s = 0.0f, sm = 0.0f;
  for (int t = 0; t < THIST; ++t) {
    const float m = mask[p * THIST + t];
    s = fmaf(h2[(size_t)(p * THIST + t) * DH + d], m, s);
    sm += m;
  }
  hist[(size_t)p * DH + d] = s / fmaxf(sm, 1.0f);
}

__global__ void k_ext_last(const float* __restrict__ past, float* __restrict__ extL) {
  const int i = blockIdx.x * blockDim.x + threadIdx.x;
  if (i >= ROWS * 2) return;
  const int r = i >> 1, c = i & 1;
  extL[i] = past[(r % P_PED) * THIST * 2 + (THIST - 1) * 2 + c];
}

__global__ void k_build_x1(const float* __restrict__ hist, const float* __restrict__ z,
                           float* __restrict__ X) {
  const int i = blockIdx.x * blockDim.x + threadIdx.x;
  const int W = DH + ZDIM;
  if (i >= ROWS * W) return;
  const int r = i / W, c = i % W;
  X[i] = (c < DH) ? hist[(size_t)(r % P_PED) * DH + c] : z[(size_t)r * ZDIM + (c - DH)];
}

__global__ void k_build_x2(const float* __restrict__ feat, const float* __restrict__ z,
                           float* __restrict__ X) {
  const int i = blockIdx.x * blockDim.x + threadIdx.x;
  const int W = F2DIM + ZDIM;
  if (i >= ROWS * W) return;
  const int r = i / W, c = i % W;
  X[i] = (c < F2DIM) ? feat[(size_t)r * F2DIM + c] : z[(size_t)r * ZDIM + (c - F2DIM)];
}

__global__ void k_coords(const float* __restrict__ xt1, const float* __restrict__ extL,
                         float* __restrict__ coords) {
  const int i = blockIdx.x * blockDim.x + threadIdx.x;
  if (i >= P_PED * 2) return;
  const int p = i >> 1, c = i & 1;
  coords[i] = xt1[(size_t)p * 2 * STRIDEK + (STRIDEK - 1) * 2 + c] + extL[p * 2 + c];
}

__global__ void k_axpy(float* __restrict__ x, const float* __restrict__ d, int n) {
  const int i = blockIdx.x * blockDim.x + threadIdx.x;
  if (i < n) x[i] += d[i];
}

__global__ void k_write_pred(const float* __restrict__ xt1, float* __restrict__ out,
                             int stage) {
  const int i = blockIdx.x * blockDim.x + threadIdx.x;
  if (i >= ROWS * 2 * STRIDEK) return;
  const int r = i / (2 * STRIDEK), kk = i % (2 * STRIDEK);
  out[(size_t)r * (2 * FUT) + stage * (2 * STRIDEK) + kk] = xt1[i];
}

__global__ void k_scene_ids(const int* __restrict__ peds, int* __restrict__ sid) {
  if (threadIdx.x | blockIdx.x) return;
  int idx = 0;
  for (int s = 0; s < NSCENES && idx < P_PED; ++s)
    for (int c = 0; c < peds[s] && idx < P_PED; ++c) sid[idx++] = s;
  while (idx < P_PED) sid[idx++] = NSCENES - 1;
}

// edge potential: one wave (32 lanes) per node i; lanes sweep candidate j
__global__ __launch_bounds__(32) void k_edge(
    const float* __restrict__ coords, const int* __restrict__ sid,
    const float* __restrict__ W1, const float* __restrict__ B1,
    const float* __restrict__ W2, const float* __restrict__ B2,
    const float* __restrict__ W3, const float* __restrict__ B3,
    float* __restrict__ feat) {
  __shared__ float sw1[48], sb1[16], sw2[16 * 32], sb2[32], sw3[32 * 32], sb3[32];
  __shared__ float sred[32][33];
  const int lane = threadIdx.x;
  for (int i = lane; i < 48; i += 32)   sw1[i] = W1[i];
  if (lane < 16)                        sb1[lane] = B1[lane];
  for (int i = lane; i < 512; i += 32)  sw2[i] = W2[i];
  sb2[lane] = B2[lane];
  for (int i = lane; i < 1024; i += 32) sw3[i] = W3[i];
  sb3[lane] = B3[lane];
  __syncthreads();

  const int i = blockIdx.x;
  const int si = sid[i];
  const float cx = coords[i * 2], cy = coords[i * 2 + 1];
  float acc[32];
#pragma unroll
  for (int d = 0; d < 32; ++d) acc[d] = 0.0f;

  for (int j0 = 0; j0 < P_PED; j0 += 32) {
    const int j = j0 + lane;
    const float dx0 = coords[j * 2] - cx, dy0 = coords[j * 2 + 1] - cy;
    const float sq = dx0 * dx0 + dy0 * dy0;
    const bool valid = (sid[j] == si) && (sq > 1e-8f) && (sq <= 9.0f);
    if (__ballot(valid) == 0ull) continue;
    const float inv = valid ? (1.0f / sqrtf(sq)) : 0.0f;
    const float dx = valid ? dx0 : 0.0f, dy = valid ? dy0 : 0.0f;
    float h1[16];
#pragma unroll
    for (int o = 0; o < 16; ++o)
      h1[o] = fmaxf(0.0f, fmaf(dx, sw1[o], fmaf(dy, sw1[16 + o],
                               fmaf(inv, sw1[32 + o], sb1[o]))));
    float h2[32];
#pragma unroll
    for (int o = 0; o < 32; ++o) {
      float a = sb2[o];
      for (int u = 0; u < 16; ++u) a = fmaf(h1[u], sw2[u * 32 + o], a);
      h2[o] = fmaxf(0.0f, a);
    }
#pragma unroll
    for (int o = 0; o < 32; ++o) {
      float a = sb3[o];
      for (int u = 0; u < 32; ++u) a = fmaf(h2[u], sw3[u * 32 + o], a);
      if (valid) acc[o] += a;
    }
  }
#pragma unroll
  for (int d = 0; d < 32; ++d) sred[lane][d] = acc[d];
  __syncthreads();
  float v = 0.0f;
  for (int l = 0; l < 32; ++l) v += sred[l][lane];
  feat[(size_t)i * F2DIM + lane] += v;
}

// ---------------- host orchestration ---------------------------------------
extern "C" void kernel_launch(void* const* d_in, const int* in_sizes, int n_in,
                              void* d_out, int out_size, void* d_ws, size_t ws_size,
                              hipStream_t stream) {
  (void)in_sizes; (void)n_in; (void)out_size; (void)ws_size;
  // setup_inputs() order, params flattened in dict/list insertion order:
  const float* past = (const float*)d_in[0];    // [768,8,2]
  const float* mask = (const float*)d_in[1];    // [768,8]
  const int*   peds = (const int*)d_in[2];      // [24]
  const float* embW = (const float*)d_in[3];    // [2,256]
  const float* embB = (const float*)d_in[4];    // [256]
  const float* Wq   = (const float*)d_in[5];
  const float* Wk   = (const float*)d_in[6];
  const float* Wv   = (const float*)d_in[7];
  const float* Wo   = (const float*)d_in[8];
  const float* uW0 = (const float*)d_in[9],  *uB0 = (const float*)d_in[10];
  const float* uW1 = (const float*)d_in[11], *uB1 = (const float*)d_in[12];
  const float* uW2 = (const float*)d_in[13], *uB2 = (const float*)d_in[14];
  const float* uW3 = (const float*)d_in[15], *uB3 = (const float*)d_in[16];
  const float* sW0 = (const float*)d_in[17], *sB0 = (const float*)d_in[18];
  const float* sW1 = (const float*)d_in[19], *sB1 = (const float*)d_in[20];
  const float* sW2 = (const float*)d_in[21], *sB2 = (const float*)d_in[22];
  const float* dW0 = (const float*)d_in[23], *dB0 = (const float*)d_in[24];
  const float* dW1 = (const float*)d_in[25], *dB1 = (const float*)d_in[26];
  const float* dW2 = (const float*)d_in[27], *dB2 = (const float*)d_in[28];
  const float* dW3 = (const float*)d_in[29], *dB3 = (const float*)d_in[30];
  const float* eW1 = (const float*)d_in[31], *eB1 = (const float*)d_in[32];
  const float* eW2 = (const float*)d_in[33], *eB2 = (const float*)d_in[34];
  const float* eW3 = (const float*)d_in[35], *eB3 = (const float*)d_in[36];
  float* out = (float*)d_out;

  // ---- workspace carve (floats) ----
  float* ws = (float*)d_ws;
  size_t off = 0;
  auto take = [&](size_t n) { size_t r = off; off += (n + 63) & ~(size_t)63; return ws + r; };
  float* hist  = take((size_t)P_PED * DH);
  float* extL  = take((size_t)ROWS * 2);
  float* z     = take((size_t)ROWS * ZDIM);
  float* xt1   = take((size_t)ROWS * 2 * STRIDEK);
  float* feat  = take((size_t)ROWS * F2DIM);
  float* f1    = take((size_t)ROWS * F2DIM);
  float* f2    = take((size_t)ROWS * F2DIM);
  float* X2    = take((size_t)ROWS * (F2DIM + ZDIM));
  float* delta = take((size_t)ROWS * 2 * STRIDEK);
  float* crd   = take((size_t)P_PED * 2);
  int*   sid   = (int*)take((size_t)P_PED);
  float* arena = take((size_t)ROWS * 272 + 2ull * ROWS * 512);  // 19.9M floats
  // phase-1 aliases (6144*256 = 1572864 each)
  const size_t HB = (size_t)6144 * DH;
  float* h0 = arena;
  float* qb = arena + HB;
  float* kb = arena + 2 * HB;
  float* vb = arena + 3 * HB;
  float* at = arena + 4 * HB;
  float* h2 = qb;                               // overwrite q after attention
  // phase-2 aliases
  float* X1 = arena;
  float* t1 = arena + (size_t)ROWS * 272;
  float* t2 = t1 + (size_t)ROWS * 512;
  float* t3 = X1;                               // X1 dead by then
  float* dd1 = arena;
  float* dd2 = arena + (size_t)ROWS * 128;
  float* dd3 = arena + 2ull * ROWS * 128;

  auto gemm = [&](const float* A, int lda, const float* B, int ldb,
                  const float* bias, const float* res, float* C, int ldc,
                  int M, int N, int K, int relu) {
    dim3 g((N + 63) / 64, M / 64);
    k_gemm<<<g, 128, 0, stream>>>(A, lda, B, ldb, bias, res, C, ldc, M, N, K, relu);
  };
  auto grid1 = [](int n, int b) { return dim3((n + b - 1) / b); };

  // ---- encoder ----
  gemm(past, 2, embW, DH, embB, nullptr, h0, DH, P_PED * THIST, DH, 2, 0);
  gemm(h0, DH, Wq, DH, nullptr, nullptr, qb, DH, P_PED * THIST, DH, DH, 0);
  gemm(h0, DH, Wk, DH, nullptr, nullptr, kb, DH, P_PED * THIST, DH, DH, 0);
  gemm(h0, DH, Wv, DH, nullptr, nullptr, vb, DH, P_PED * THIST, DH, DH, 0);
  k_attn<<<P_PED, 64, 0, stream>>>(qb, kb, vb, mask, at);
  gemm(at, DH, Wo, DH, nullptr, h0, h2, DH, P_PED * THIST, DH, DH, 0);
  k_pool<<<P_PED, DH, 0, stream>>>(h2, mask, hist);
  k_ext_last<<<grid1(ROWS * 2, 256), 256, 0, stream>>>(past, extL);

  // ---- RNG key chain (host-side JAX threefry split chain, seed 42) ----
  uint32_t key[2] = {0u, 42u}, sub[2];
  host_split_(key, sub);                        // z1 key
  k_normal<<<grid1(ROWS * ZDIM / 2, 256), 256, 0, stream>>>(
      sub[0], sub[1], ROWS * ZDIM, 1.0f /*SIGMA1*/, z);

  // ---- update_decoder: [272 -> 512 -> 512 -> 256 -> 10] ----
  k_build_x1<<<grid1(ROWS * (DH + ZDIM), 256), 256, 0, stream>>>(hist, z, X1);
  gemm(X1, 272, uW0, 512, uB0, nullptr, t1, 512, ROWS, 512, 272, 1);
  gemm(t1, 512, uW1, 512, uB1, nullptr, t2, 512, ROWS, 512, 512, 1);
  gemm(t2, 512, uW2, 256, uB2, nullptr, t3, 256, ROWS, 256, 512, 1);
  gemm(t3, 256, uW3, 10, uB3, nullptr, xt1, 10, ROWS, 10, 256, 0);
  k_write_pred<<<grid1(ROWS * 10, 256), 256, 0, stream>>>(xt1, out, 0);

  k_scene_ids<<<1, 32, 0, stream>>>(peds, sid);
  k_coords<<<grid1(P_PED * 2, 256), 256, 0, stream>>>(xt1, extL, crd);

  // ---- 3 refinement iterations ----
  for (int stage = 1; stage <= 3; ++stage) {
    gemm(xt1, 10, sW0, 32, sB0, nullptr, f1, 32, ROWS, 32, 10, 1);
    gemm(f1, 32, sW1, 32, sB1, nullptr, f2, 32, ROWS, 32, 32, 1);
    gemm(f2, 32, sW2, 32, sB2, nullptr, feat, 32, ROWS, 32, 32, 0);
    k_edge<<<P_PED, 32, 0, stream>>>(crd, sid, eW1, eB1, eW2, eB2, eW3, eB3, feat);

    host_split_(key, sub);                      // z2 key for this iteration
    k_normal<<<grid1(ROWS * ZDIM / 2, 256), 256, 0, stream>>>(
        sub[0], sub[1], ROWS * ZDIM, 1.0f /*SIGMA2*/, z);
    k_build_x2<<<grid1(ROWS * (F2DIM + ZDIM), 256), 256, 0, stream>>>(feat, z, X2);

    gemm(X2, 48, dW0, 128, dB0, nullptr, dd1, 128, ROWS, 128, 48, 1);
    gemm(dd1, 128, dW1, 128, dB1, nullptr, dd2, 128, ROWS, 128, 128, 1);
    gemm(dd2, 128, dW2, 64, dB2, nullptr, dd3, 64, ROWS, 64, 128, 1);
    gemm(dd3, 64, dW3, 10, dB3, nullptr, delta, 10, ROWS, 10, 64, 0);

    k_axpy<<<grid1(ROWS * 10, 256), 256, 0, stream>>>(xt1, delta, ROWS * 10);
    k_write_pred<<<grid1(ROWS * 10, 256), 256, 0, stream>>>(xt1, out, stage);
    k_coords<<<grid1(P_PED * 2, 256), 256, 0, stream>>>(xt1, extL, crd);
  }
}